// HilbertFlatten_13400297963779
// MI455X (gfx1250) — compile-verified
//
#include <hip/hip_runtime.h>
#include <stdint.h>

// ---------------------------------------------------------------------------
// HilbertFlatten for a 256x256x256 f32 volume, num_bits = 8, num_dims = 3.
// out[s] = x.flat[ decode(s) ], s = 0 .. 2^24-1 (Skilling decode, matching
// numpy-hilbert-curve / the reference's vectorized loop exactly).
//
// Each thread owns one aligned leaf octant (8 consecutive Hilbert indices):
//   - full branchless Skilling transform for planes 7..1 (base decode)
//   - plane 0 tracked symbolically as (perm,flip) per output dim, since the
//     reference's transforms at plane q read only ORIGINAL gray bits at plane
//     q and write only planes < q  =>  plane-0 result is an affine map of the
//     leaf gray bits, identical across the 8 offsets.
// Memory: 8x global_load_b32 gather (L2-resident: 64MB input << 192MB L2),
//         2x non-temporal global_store_b128 coalesced stream.
// ---------------------------------------------------------------------------

typedef __attribute__((ext_vector_type(4))) float v4f;

__device__ __forceinline__ uint32_t compact3(uint32_t x) {
  // Keep bits 0,3,6,...,21 and compact to bits 0..7 (inverse Morton part-by-2)
  x &= 0x09249249u;
  x = (x ^ (x >> 2))  & 0x030C30C3u;
  x = (x ^ (x >> 4))  & 0x0300F00Fu;
  x = (x ^ (x >> 8))  & 0xFF0000FFu;
  x = (x ^ (x >> 16)) & 0x000003FFu;
  return x;
}

__global__ __launch_bounds__(256) void hilbert_flatten_f32(
    const float* __restrict__ x, float* __restrict__ out, uint32_t n_groups) {
  const uint32_t group = blockIdx.x * blockDim.x + threadIdx.x;
  if (group >= n_groups) return;

  const uint32_t h = group << 3;        // base Hilbert index, low 3 bits = 0
  const uint32_t g = h ^ (h >> 1);      // 24-bit gray code

  // De-interleave: X[d] bit k = g bit (3k + 2 - d). Clear plane 0 (symbolic).
  uint32_t X0 = compact3(g >> 2) & 0xFEu;
  uint32_t X1 = compact3(g >> 1) & 0xFEu;
  uint32_t X2 = compact3(g)      & 0xFEu;

  // Plane-0 symbolic state per output register r: pf_r = (perm_r<<1)|flip_r,
  // final plane-0 bit of reg r = leafgray[perm_r] ^ flip_r.
  uint32_t pf0 = 0u, pf1 = 2u, pf2 = 4u;

#pragma unroll
  for (uint32_t q = 1u; q < 8u; ++q) {
    const uint32_t low = (1u << q) - 2u;          // bits [1, q-1] (plane 0 excluded)
    {   // dim = 2
      const bool     c  = ((X2 >> q) & 1u) != 0u; // original gray bit (planes>=q untouched)
      const uint32_t sw = (X0 ^ X2) & low;
      X0 ^= c ? low : sw;                         // invert X0 low bits : swap with X2
      X2 ^= c ? 0u  : sw;
      const uint32_t n0 = c ? (pf0 ^ 1u) : pf2;   // plane-0: flip f0 : swap pf0<->pf2
      const uint32_t n2 = c ? pf2        : pf0;
      pf0 = n0; pf2 = n2;
    }
    {   // dim = 1
      const bool     c  = ((X1 >> q) & 1u) != 0u;
      const uint32_t sw = (X0 ^ X1) & low;
      X0 ^= c ? low : sw;
      X1 ^= c ? 0u  : sw;
      const uint32_t n0 = c ? (pf0 ^ 1u) : pf1;
      const uint32_t n1 = c ? pf1        : pf0;
      pf0 = n0; pf1 = n1;
    }
    {   // dim = 0 (swap with self is a no-op; only the conditional invert acts)
      const uint32_t t = (X0 >> q) & 1u;
      X0 ^= t ? low : 0u;
      pf0 ^= t;
    }
  }

  const uint32_t base_flat = (X0 << 16) | (X1 << 8) | X2;   // plane-0 bits are 0 here

  // Leaf gray triples for offsets o=0..7, packed one nibble each:
  // w(o) bit d = gray bit (2-d) of (h|o); only h bit 3 leaks in (flips bit 0).
  uint32_t W = 0x15732640u ^ ((h & 8u) ? 0x11111111u : 0u);
  // Pre-apply perm (shift) and flip (xor) so the leaf loop is immediate-shift only.
  const uint32_t W0 = (W >> (pf0 >> 1)) ^ (0u - (pf0 & 1u));
  const uint32_t W1 = (W >> (pf1 >> 1)) ^ (0u - (pf1 & 1u));
  const uint32_t W2 = (W >> (pf2 >> 1)) ^ (0u - (pf2 & 1u));

  float v[8];
#pragma unroll
  for (int o = 0; o < 8; ++o) {
    const uint32_t flat = base_flat
        | (((W0 >> (4 * o)) & 1u) << 16)
        | (((W1 >> (4 * o)) & 1u) << 8)
        |  ((W2 >> (4 * o)) & 1u);
    v[o] = x[flat];                      // gather: spatially local within the wave
  }

  // Coalesced 16B-aligned streaming stores; NT keeps the write stream from
  // evicting the L2-resident input across graph replays.
  v4f lo = {v[0], v[1], v[2], v[3]};
  v4f hi = {v[4], v[5], v[6], v[7]};
  v4f* dst = reinterpret_cast<v4f*>(out + (size_t)h);
  __builtin_nontemporal_store(lo, dst);
  __builtin_nontemporal_store(hi, dst + 1);
}

extern "C" void kernel_launch(void* const* d_in, const int* in_sizes, int n_in,
                              void* d_out, int out_size, void* d_ws, size_t ws_size,
                              hipStream_t stream) {
  (void)in_sizes; (void)n_in; (void)d_ws; (void)ws_size;
  const float* x = reinterpret_cast<const float*>(d_in[0]);
  float* out = reinterpret_cast<float*>(d_out);

  const uint32_t n = (uint32_t)out_size;      // 16777216 = 2^24
  const uint32_t n_groups = n >> 3;           // one thread per 8 outputs
  const uint32_t block = 256;                 // 8 wave32 waves per workgroup
  const uint32_t grid = (n_groups + block - 1) / block;

  hilbert_flatten_f32<<<grid, block, 0, stream>>>(x, out, n_groups);
}